// FixedRandomMap_51951924413084
// MI455X (gfx1250) — compile-verified
//
#include <hip/hip_runtime.h>

typedef float v2f __attribute__((ext_vector_type(2)));
typedef float v8f __attribute__((ext_vector_type(8)));
typedef int   v4i __attribute__((vector_size(16)));   // matches builtin's int4

#define DSZ    1024
#define TILE_M 128
#define TILE_N 128
#define TILE_K 32
#define NT     (DSZ / TILE_K)   // 32 K-chunks
#define LDA    36    // padded row stride of As in floats (144 B, 16B aligned)
#define LDB    144   // padded row stride of Bs in floats (576 B, 16B aligned)

#define AS1 __attribute__((address_space(1)))
#define AS3 __attribute__((address_space(3)))

#if defined(__has_builtin)
#if __has_builtin(__builtin_amdgcn_global_load_async_to_lds_b128)
#define HAVE_ASYNC_LDS 1
#endif
#if __has_builtin(__builtin_amdgcn_s_wait_asynccnt)
#define HAVE_WAIT_ASYNC 1
#endif
#endif
#ifndef HAVE_ASYNC_LDS
#define HAVE_ASYNC_LDS 0
#endif
#ifndef HAVE_WAIT_ASYNC
#define HAVE_WAIT_ASYNC 0
#endif

__device__ __forceinline__ void copy_b128_to_lds(const float* g, float* l) {
#if HAVE_ASYNC_LDS
    // GLOBAL_LOAD_ASYNC_TO_LDS_B128: per-lane 16B, memory -> LDS, ASYNCcnt.
    __builtin_amdgcn_global_load_async_to_lds_b128(
        (AS1 v4i*)g, (AS3 v4i*)l, /*offset=*/0, /*cpol=*/0);
#else
    *(float4*)l = *(const float4*)g;   // sync fallback: VGPR-staged
#endif
}

__device__ __forceinline__ void wait_lds_copies() {
#if HAVE_ASYNC_LDS
#if HAVE_WAIT_ASYNC
    __builtin_amdgcn_s_wait_asynccnt(0);
#else
    asm volatile("s_wait_asynccnt 0x0" ::: "memory");
#endif
#endif
}

// out[b] = T[fwd_steps[b]-1] @ x[b], all 1024x1024 fp32.
__global__ __launch_bounds__(256)
void gather_bmm_f32_wmma(const float* __restrict__ x,
                         const int* __restrict__ fwd_steps,
                         const float* __restrict__ T,
                         float* __restrict__ out)
{
    __shared__ float As[2][TILE_M * LDA];   // As[buf][row][kk]   (A = T[tb] tile)
    __shared__ float Bs[2][TILE_K * LDB];   // Bs[buf][kk][col]   (B = x[b]  tile)

    const int b  = blockIdx.z;
    const int m0 = blockIdx.y * TILE_M;
    const int n0 = blockIdx.x * TILE_N;

    const int t    = threadIdx.x;
    const int lane = t & 31;
    const int wave = t >> 5;
    const int wm   = wave >> 1;   // 0..3 : 32-row slab
    const int wn   = wave & 1;    // 0..1 : 64-col slab
    const int l16  = lane & 15;
    const int hi   = lane >> 4;   // 0/1 : K pair select (and M+8 for C/D)

    const int tb = fwd_steps[b] - 1;
    const float* __restrict__ Am = T + (size_t)tb * DSZ * DSZ;
    const float* __restrict__ Bm = x + (size_t)b  * DSZ * DSZ;
    float* __restrict__ Cm = out + (size_t)b * DSZ * DSZ;

    v8f acc[2][4];
#pragma unroll
    for (int i = 0; i < 2; ++i)
#pragma unroll
        for (int j = 0; j < 4; ++j)
            acc[i][j] = (v8f){0.f, 0.f, 0.f, 0.f, 0.f, 0.f, 0.f, 0.f};

    // A-tile: 8 threads x 16B per 32-float row, 4 passes of 32 rows.
    const int ar = t >> 3;          // 0..31
    const int ac = (t & 7) << 2;    // 0,4,..,28
    // B-tile: 32 threads x 16B per 128-float row, 4 passes of 8 rows.
    const int br = t >> 5;          // 0..7
    const int bc = (t & 31) << 2;   // 0,4,..,124

    // Issue this thread's 8 async 16B copies for K-chunk k0 into buffer `buf`.
    auto stageTileAsync = [&](int buf, int k0) {
#pragma unroll
        for (int p = 0; p < 4; ++p)
            copy_b128_to_lds(Am + (size_t)(m0 + ar + p * 32) * DSZ + k0 + ac,
                             &As[buf][(ar + p * 32) * LDA + ac]);
#pragma unroll
        for (int p = 0; p < 4; ++p)
            copy_b128_to_lds(Bm + (size_t)(k0 + br + p * 8) * DSZ + n0 + bc,
                             &Bs[buf][(br + p * 8) * LDB + bc]);
    };

    auto computeTile = [&](int buf) {
#pragma unroll
        for (int s = 0; s < 8; ++s) {
            const int kb = s * 4 + hi * 2;   // this lane's K pair within chunk

            v2f a[2];
#pragma unroll
            for (int i = 0; i < 2; ++i) {
                const float* p = &As[buf][(wm * 32 + i * 16 + l16) * LDA + kb];
                a[i] = *(const v2f*)p;        // 8B aligned: kb even, LDA even
            }

            v2f bf[4];
#pragma unroll
            for (int j = 0; j < 4; ++j) {
                const int c = wn * 64 + j * 16 + l16;
                bf[j].x = Bs[buf][kb * LDB + c];
                bf[j].y = Bs[buf][(kb + 1) * LDB + c];
            }

#pragma unroll
            for (int i = 0; i < 2; ++i)
#pragma unroll
                for (int j = 0; j < 4; ++j)
                    acc[i][j] = __builtin_amdgcn_wmma_f32_16x16x4_f32(
                        false, a[i], false, bf[j],
                        (short)0, acc[i][j], false, false);
        }
    };

    // ---- software-pipelined main loop: async staging, 1 barrier per chunk ----
    stageTileAsync(0, 0);
    wait_lds_copies();
    __syncthreads();

    for (int i = 0; i < NT - 1; ++i) {
        stageTileAsync((i + 1) & 1, (i + 1) * TILE_K);  // fills other buffer
        computeTile(i & 1);                             // overlaps async copies
        wait_lds_copies();
        __syncthreads();
    }
    computeTile((NT - 1) & 1);

    // ---- store D fragments: VGPR v -> row (v + 8*hi), lane%16 -> col ----
    // Output is streamed once: nontemporal so it doesn't pollute L2 (keep T set hot).
#pragma unroll
    for (int i = 0; i < 2; ++i) {
#pragma unroll
        for (int j = 0; j < 4; ++j) {
            const int col = n0 + wn * 64 + j * 16 + l16;
            const int rbase = m0 + wm * 32 + i * 16 + hi * 8;
#pragma unroll
            for (int v = 0; v < 8; ++v) {
                __builtin_nontemporal_store(acc[i][j][v],
                                            Cm + (size_t)(rbase + v) * DSZ + col);
            }
        }
    }
}

extern "C" void kernel_launch(void* const* d_in, const int* in_sizes, int n_in,
                              void* d_out, int out_size, void* d_ws, size_t ws_size,
                              hipStream_t stream) {
    const float* x   = (const float*)d_in[0];
    const int*   fwd = (const int*)d_in[1];   // JAX default: int32 (x64 disabled)
    const float* T   = (const float*)d_in[2];
    float*       out = (float*)d_out;

    const int nb = in_sizes[1];               // batch count (128)
    dim3 grid(DSZ / TILE_N, DSZ / TILE_M, nb);
    gather_bmm_f32_wmma<<<grid, 256, 0, stream>>>(x, fwd, T, out);
}